// TGATModel_47742856462867
// MI455X (gfx1250) — compile-verified
//
#include <hip/hip_runtime.h>
#include <math.h>

// ---- problem constants (match reference) ----
constexpr int T_  = 8;
constexpr int N_  = 20000;
constexpr int E_  = 640000;
constexpr int ET_ = E_ + N_;   // edges + self loops
constexpr int H_  = 32;        // hidden per head
constexpr int D1_ = 64;        // heads*H (layer1 out)
constexpr int G_  = 96;        // 3*H (GRU gates)

typedef __attribute__((ext_vector_type(2))) float v2f;
typedef __attribute__((ext_vector_type(8))) float v8f;

// ---------- helpers ----------
__device__ __forceinline__ unsigned f2sort(float f) {
  unsigned u = __float_as_uint(f);
  return u ^ ((u >> 31) ? 0xFFFFFFFFu : 0x80000000u);
}
__device__ __forceinline__ float sort2f(unsigned u) {
  unsigned b = (u & 0x80000000u) ? (u ^ 0x80000000u) : ~u;
  return __uint_as_float(b);
}
__device__ __forceinline__ float lrelu02(float v) { return v > 0.f ? v : 0.2f * v; }
__device__ __forceinline__ float sigmoidf(float v) { return 1.0f / (1.0f + expf(-v)); }

// ---------- GAT layer 1 per-node: h1 = x*W1, alphas, and per-step init ----------
__global__ void k_gat1_node(const float* __restrict__ x_t, const float* __restrict__ W1,
                            const float* __restrict__ a_s, const float* __restrict__ a_d,
                            float* __restrict__ h1, float* __restrict__ as,
                            float* __restrict__ ad, float* __restrict__ agg,
                            unsigned* __restrict__ m, float* __restrict__ den) {
  int n = blockIdx.x * blockDim.x + threadIdx.x;
  if (n >= N_) return;
  float xv = x_t[n];
  float s0 = 0.f, s1 = 0.f, d0 = 0.f, d1 = 0.f;
  #pragma unroll 8
  for (int c = 0; c < D1_; ++c) {
    float hv = xv * W1[c];
    h1[(size_t)n * D1_ + c]  = hv;
    agg[(size_t)n * D1_ + c] = 0.f;
    float as_w = a_s[c], ad_w = a_d[c];   // a_src1 is [2][32] flat == [64]
    if (c < H_) { s0 += hv * as_w; d0 += hv * ad_w; }
    else        { s1 += hv * as_w; d1 += hv * ad_w; }
  }
  as[n * 2] = s0; as[n * 2 + 1] = s1;
  ad[n * 2] = d0; ad[n * 2 + 1] = d1;
  m[n * 2] = 0u; m[n * 2 + 1] = 0u;       // sortable(-inf) > 0, any finite beats 0
  den[n * 2] = 0.f; den[n * 2 + 1] = 0.f;
}

// ---------- GAT layer 2 per-node: alphas from h2, init agg/m/den ----------
__global__ void k_gat2_node(const float* __restrict__ h2, const float* __restrict__ a_s,
                            const float* __restrict__ a_d, float* __restrict__ as,
                            float* __restrict__ ad, float* __restrict__ agg,
                            unsigned* __restrict__ m, float* __restrict__ den) {
  int n = blockIdx.x * blockDim.x + threadIdx.x;
  if (n >= N_) return;
  float s = 0.f, d = 0.f;
  #pragma unroll 8
  for (int c = 0; c < H_; ++c) {
    float hv = h2[(size_t)n * H_ + c];
    s += hv * a_s[c]; d += hv * a_d[c];
    agg[(size_t)n * H_ + c] = 0.f;
  }
  as[n] = s; ad[n] = d; m[n] = 0u; den[n] = 0.f;
}

// ---------- edge pass 1: segment-max of leaky_relu(as[src]+ad[dst]) ----------
template <int NH>
__global__ void k_edge_max(const int* __restrict__ src_a, const int* __restrict__ dst_a,
                           const float* __restrict__ as, const float* __restrict__ ad,
                           unsigned* __restrict__ m) {
  int e = blockIdx.x * blockDim.x + threadIdx.x;
  if (e >= ET_) return;
  int s = (e < E_) ? src_a[e] : (e - E_);
  int d = (e < E_) ? dst_a[e] : (e - E_);
  #pragma unroll
  for (int h = 0; h < NH; ++h) {
    float v = lrelu02(as[s * NH + h] + ad[d * NH + h]);
    atomicMax(&m[d * NH + h], f2sort(v));
  }
}

// ---------- edge pass 2: w = exp(e - max); denom += w; stash w ----------
template <int NH>
__global__ void k_edge_exp(const int* __restrict__ src_a, const int* __restrict__ dst_a,
                           const float* __restrict__ as, const float* __restrict__ ad,
                           const unsigned* __restrict__ m, float* __restrict__ den,
                           float* __restrict__ wbuf) {
  int e = blockIdx.x * blockDim.x + threadIdx.x;
  if (e >= ET_) return;
  int s = (e < E_) ? src_a[e] : (e - E_);
  int d = (e < E_) ? dst_a[e] : (e - E_);
  #pragma unroll
  for (int h = 0; h < NH; ++h) {
    float v = lrelu02(as[s * NH + h] + ad[d * NH + h]);
    float w = expf(v - sort2f(m[d * NH + h]));
    wbuf[(size_t)e * NH + h] = w;
    atomicAdd(&den[d * NH + h], w);
  }
}

// ---------- edge pass 3: agg[dst] += h[src] * w/(den+eps) ----------
template <int NH, int C>
__global__ void k_edge_agg(const int* __restrict__ src_a, const int* __restrict__ dst_a,
                           const float* __restrict__ hfeat, const float* __restrict__ wbuf,
                           const float* __restrict__ den, float* __restrict__ agg) {
  int e = blockIdx.x * blockDim.x + threadIdx.x;
  if (e >= ET_) return;
  int s = (e < E_) ? src_a[e] : (e - E_);
  int d = (e < E_) ? dst_a[e] : (e - E_);
  float w[NH];
  #pragma unroll
  for (int h = 0; h < NH; ++h)
    w[h] = wbuf[(size_t)e * NH + h] / (den[d * NH + h] + 1e-16f);
  constexpr int D = NH * C;
  const float* hs = hfeat + (size_t)s * D;
  float* ag = agg + (size_t)d * D;
  #pragma unroll 8
  for (int c = 0; c < D; ++c)
    atomicAdd(&ag[c], hs[c] * w[c / C]);
}

// ---------- finalize: z = elu(agg + bias), in place ----------
__global__ void k_elu_bias(float* __restrict__ z, const float* __restrict__ bias,
                           int dmask, int total) {
  int i = blockIdx.x * blockDim.x + threadIdx.x;
  if (i >= total) return;
  float v = z[i] + bias[i & dmask];
  z[i] = v > 0.f ? v : (expf(v) - 1.f);
}

// ---------- dense GEMM via V_WMMA_F32_16X16X4_F32 ----------
// C[n,M] = A[n,K] @ B[K,M] (+bias). One wave per 16x16 tile. Requires n%16==0,
// K%4==0, M%16==0. EXEC all-ones within active waves (wave-uniform exit only).
// VGPR layouts per CDNA5 ISA 7.12.2:
//   A 16x4 : lane holds {A[m][kl], A[m][kl+1]}, m=lane&15, kl=(lane>>4)*2
//   B 4x16 : lane holds {B[kl][col], B[kl+1][col]}, col=lane&15
//   C 16x16: VGPR i -> row = i + (lane>=16 ? 8 : 0), col = lane&15
__global__ void k_wmma_mm(const float* __restrict__ A, const float* __restrict__ B,
                          const float* __restrict__ bias, float* __restrict__ C,
                          int K, int M) {
  int wave = (blockIdx.x * blockDim.x + threadIdx.x) >> 5;
  int lane = threadIdx.x & 31;
  int colTiles = M >> 4;
  int nTiles = (N_ >> 4) * colTiles;
  if (wave >= nTiles) return;                 // wave-uniform
  int row0 = (wave / colTiles) << 4;
  int col0 = (wave % colTiles) << 4;
  int mrow = lane & 15;
  int kl   = (lane >> 4) << 1;                // 0 or 2
  const float* Arow = A + (size_t)(row0 + mrow) * K;
  int col = col0 + mrow;
  v8f acc = {};
  for (int k0 = 0; k0 < K; k0 += 4) {
    v2f a, b;
    a.x = Arow[k0 + kl];
    a.y = Arow[k0 + kl + 1];
    b.x = B[(size_t)(k0 + kl) * M + col];
    b.y = B[(size_t)(k0 + kl + 1) * M + col];
    acc = __builtin_amdgcn_wmma_f32_16x16x4_f32(false, a, false, b,
                                                (short)0, acc, false, false);
  }
  float bv = bias ? bias[col] : 0.f;
  int rbase = row0 + ((lane >> 4) << 3);
  #pragma unroll
  for (int i = 0; i < 8; ++i)
    C[(size_t)(rbase + i) * M + col] = acc[i] + bv;
}

// ---------- GRU gates (torch order r,z,n), h updated in place ----------
__global__ void k_gru_gate(const float* __restrict__ gi, const float* __restrict__ gh,
                           float* __restrict__ h) {
  int i = blockIdx.x * blockDim.x + threadIdx.x;
  if (i >= N_ * H_) return;
  int n = i >> 5, f = i & 31;
  const float* gin = gi + (size_t)n * G_;
  const float* ghn = gh + (size_t)n * G_;
  float r  = sigmoidf(gin[f] + ghn[f]);
  float zz = sigmoidf(gin[H_ + f] + ghn[H_ + f]);
  float ng = tanhf(gin[2 * H_ + f] + r * ghn[2 * H_ + f]);
  h[i] = (1.f - zz) * ng + zz * h[i];
}

// ---------- weight transpose: W[96,32] -> WT[32,96] ----------
__global__ void k_transpose(const float* __restrict__ W, float* __restrict__ WT) {
  int i = blockIdx.x * blockDim.x + threadIdx.x;
  if (i >= G_ * H_) return;
  int k = i / G_, mm = i % G_;
  WT[i] = W[mm * H_ + k];
}

// ---------- output: out = h @ W_out^T + b_out ----------
__global__ void k_out(const float* __restrict__ h, const float* __restrict__ W_out,
                      const float* __restrict__ b_out, float* __restrict__ out) {
  int n = blockIdx.x * blockDim.x + threadIdx.x;
  if (n >= N_) return;
  float acc = 0.f;
  #pragma unroll 8
  for (int c = 0; c < H_; ++c) acc += h[(size_t)n * H_ + c] * W_out[c];
  out[n] = acc + b_out[0];
}

extern "C" void kernel_launch(void* const* d_in, const int* in_sizes, int n_in,
                              void* d_out, int out_size, void* d_ws, size_t ws_size,
                              hipStream_t stream) {
  const float* x    = (const float*)d_in[0];
  const int*   ei   = (const int*)  d_in[1];
  const float* W1   = (const float*)d_in[2];
  const float* aS1  = (const float*)d_in[3];
  const float* aD1  = (const float*)d_in[4];
  const float* b1   = (const float*)d_in[5];
  const float* W2   = (const float*)d_in[6];
  const float* aS2  = (const float*)d_in[7];
  const float* aD2  = (const float*)d_in[8];
  const float* b2   = (const float*)d_in[9];
  const float* W_ih = (const float*)d_in[10];
  const float* W_hh = (const float*)d_in[11];
  const float* b_ih = (const float*)d_in[12];
  const float* b_hh = (const float*)d_in[13];
  const float* Wout = (const float*)d_in[14];
  const float* bout = (const float*)d_in[15];
  float* out = (float*)d_out;

  // ---- workspace layout (floats) ----
  float* ws = (float*)d_ws;
  size_t o = 0;
  float*    h1   = ws + o; o += (size_t)N_ * D1_;
  float*    z1   = ws + o; o += (size_t)N_ * D1_;   // agg1 -> z1 in place
  float*    as1  = ws + o; o += (size_t)N_ * 2;
  float*    ad1  = ws + o; o += (size_t)N_ * 2;
  unsigned* m1   = (unsigned*)(ws + o); o += (size_t)N_ * 2;
  float*    den1 = ws + o; o += (size_t)N_ * 2;
  float*    h2   = ws + o; o += (size_t)N_ * H_;
  float*    z2   = ws + o; o += (size_t)N_ * H_;    // agg2 -> z2 in place
  float*    as2  = ws + o; o += N_;
  float*    ad2  = ws + o; o += N_;
  unsigned* m2   = (unsigned*)(ws + o); o += N_;
  float*    den2 = ws + o; o += N_;
  float*    hprv = ws + o; o += (size_t)N_ * H_;
  float*    gi   = ws + o; o += (size_t)N_ * G_;
  float*    gh   = ws + o; o += (size_t)N_ * G_;
  float*    wbuf = ws + o; o += (size_t)ET_ * 2;
  float*    WihT = ws + o; o += G_ * H_;
  float*    WhhT = ws + o; o += G_ * H_;

  const int BLK = 256;
  const int gN  = (N_ + BLK - 1) / BLK;
  const int gE  = (ET_ + BLK - 1) / BLK;

  // one-time per call: h0 = 0, pre-transpose GRU weights
  hipMemsetAsync(hprv, 0, (size_t)N_ * H_ * sizeof(float), stream);
  k_transpose<<<(G_ * H_ + BLK - 1) / BLK, BLK, 0, stream>>>(W_ih, WihT);
  k_transpose<<<(G_ * H_ + BLK - 1) / BLK, BLK, 0, stream>>>(W_hh, WhhT);

  for (int t = 0; t < T_; ++t) {
    const float* x_t = x + (size_t)t * N_;
    const int* src_a = ei + (size_t)t * 2 * E_;
    const int* dst_a = src_a + E_;

    // --- GAT layer 1 (2 heads, 32 ch/head) ---
    k_gat1_node<<<gN, BLK, 0, stream>>>(x_t, W1, aS1, aD1, h1, as1, ad1, z1, m1, den1);
    k_edge_max<2><<<gE, BLK, 0, stream>>>(src_a, dst_a, as1, ad1, m1);
    k_edge_exp<2><<<gE, BLK, 0, stream>>>(src_a, dst_a, as1, ad1, m1, den1, wbuf);
    k_edge_agg<2, H_><<<gE, BLK, 0, stream>>>(src_a, dst_a, h1, wbuf, den1, z1);
    k_elu_bias<<<((size_t)N_ * D1_ + BLK - 1) / BLK, BLK, 0, stream>>>(z1, b1, D1_ - 1, N_ * D1_);

    // --- GAT layer 2: h2 = z1 @ W2 (WMMA f32), then attention (1 head) ---
    {
      int waves = (N_ / 16) * (H_ / 16);
      k_wmma_mm<<<(waves * 32 + BLK - 1) / BLK, BLK, 0, stream>>>(z1, W2, nullptr, h2, D1_, H_);
    }
    k_gat2_node<<<gN, BLK, 0, stream>>>(h2, aS2, aD2, as2, ad2, z2, m2, den2);
    k_edge_max<1><<<gE, BLK, 0, stream>>>(src_a, dst_a, as2, ad2, m2);
    k_edge_exp<1><<<gE, BLK, 0, stream>>>(src_a, dst_a, as2, ad2, m2, den2, wbuf);
    k_edge_agg<1, H_><<<gE, BLK, 0, stream>>>(src_a, dst_a, h2, wbuf, den2, z2);
    k_elu_bias<<<((size_t)N_ * H_ + BLK - 1) / BLK, BLK, 0, stream>>>(z2, b2, H_ - 1, N_ * H_);

    // --- GRU: gi = z2 @ W_ih^T + b_ih ; gh = h @ W_hh^T + b_hh (WMMA f32) ---
    {
      int waves = (N_ / 16) * (G_ / 16);
      int grd = (waves * 32 + BLK - 1) / BLK;
      k_wmma_mm<<<grd, BLK, 0, stream>>>(z2,   WihT, b_ih, gi, H_, G_);
      k_wmma_mm<<<grd, BLK, 0, stream>>>(hprv, WhhT, b_hh, gh, H_, G_);
    }
    k_gru_gate<<<((size_t)N_ * H_ + BLK - 1) / BLK, BLK, 0, stream>>>(gi, gh, hprv);
  }

  k_out<<<gN, BLK, 0, stream>>>(hprv, Wout, bout, out);
}